// GCN_79860621902688
// MI455X (gfx1250) — compile-verified
//
#include <hip/hip_runtime.h>
#include <hip/hip_bf16.h>

typedef float v2f __attribute__((ext_vector_type(2)));
typedef float v8f __attribute__((ext_vector_type(8)));

// ---------------------------------------------------------------------------
// 1) zero the degree accumulator
// ---------------------------------------------------------------------------
__global__ void gcn_zero_deg(float* __restrict__ deg, int n) {
    int i = blockIdx.x * blockDim.x + threadIdx.x;
    if (i < n) deg[i] = 0.0f;
}

// ---------------------------------------------------------------------------
// 2) degree of each target node (float atomics -> global_atomic_add_f32)
// ---------------------------------------------------------------------------
__global__ void gcn_count_deg(const long long* __restrict__ col,
                              float* __restrict__ deg, int n_edges) {
    int e = blockIdx.x * blockDim.x + threadIdx.x;
    if (e < n_edges) {
        atomicAdd(&deg[(int)col[e]], 1.0f);
    }
}

// ---------------------------------------------------------------------------
// 3) h = x @ W via V_WMMA_F32_16X16X4_F32.
//    One wave32 computes one 16x16 output tile (16 nodes).
//    A (16x4 f32): lane L holds row (L&15), K = 2*(L>>4) + {0,1}     (2 VGPRs)
//    B (4x16 f32): lane L holds col (L&15), K = 2*(L>>4) + {0,1}     (2 VGPRs)
//    D (16x16 f32): VGPR v holds M = v + 8*(L>>4), N = (L&15)        (8 VGPRs)
//    K=16 accumulated over 4 chained WMMAs.
// ---------------------------------------------------------------------------
__global__ void gcn_h_wmma(const float* __restrict__ x,
                           const float* __restrict__ W,
                           float* __restrict__ h, int n_nodes) {
    const int wave = (blockIdx.x * blockDim.x + threadIdx.x) >> 5;
    const int lane = threadIdx.x & 31;
    const int row_base = wave << 4;          // 16 rows per wave
    if (row_base >= n_nodes) return;         // wave-uniform: EXEC stays all-1s

    const int half = lane >> 4;              // 0 or 1
    const int l16  = lane & 15;
    int rowA = row_base + l16;
    if (rowA >= n_nodes) rowA = n_nodes - 1; // safe clamp for ragged last tile

    v8f c = {};
#pragma unroll
    for (int k0 = 0; k0 < 16; k0 += 4) {
        const int kA = k0 + 2 * half;
        v2f a, b;
        a.x = x[rowA * 16 + kA + 0];
        a.y = x[rowA * 16 + kA + 1];
        b.x = W[(kA + 0) * 16 + l16];
        b.y = W[(kA + 1) * 16 + l16];
        // 8 args: (neg_a, A, neg_b, B, c_mod, C, reuse_a, reuse_b)
        c = __builtin_amdgcn_wmma_f32_16x16x4_f32(
                false, a, false, b, (short)0, c, false, false);
    }

    // Fast path: full tile -> 8 unconditional coalesced b32 stores (one clause).
    if (row_base + 16 <= n_nodes) {
#pragma unroll
        for (int v = 0; v < 8; ++v) {
            const int m = row_base + v + 8 * half;
            h[m * 16 + l16] = c[v];
        }
    } else {
        // Ragged tail (never taken when N % 16 == 0).
#pragma unroll
        for (int v = 0; v < 8; ++v) {
            const int m = row_base + v + 8 * half;
            if (m < n_nodes) h[m * 16 + l16] = c[v];
        }
    }
}

// ---------------------------------------------------------------------------
// 4) dinv = rsqrt(deg + 1)   (in place over deg buffer)
// ---------------------------------------------------------------------------
__global__ void gcn_dinv(float* __restrict__ deg, int n) {
    int i = blockIdx.x * blockDim.x + threadIdx.x;
    if (i < n) deg[i] = rsqrtf(deg[i] + 1.0f);
}

// ---------------------------------------------------------------------------
// 5) out = h * dinv^2 + b   (self-loop term + bias; initializes d_out)
//    float4 per thread: one b128 load + one b128 store.
// ---------------------------------------------------------------------------
__global__ void gcn_init_out(const float4* __restrict__ h4,
                             const float* __restrict__ dinv,
                             const float4* __restrict__ b4,
                             float4* __restrict__ out4, int n_nodes) {
    int t = blockIdx.x * blockDim.x + threadIdx.x;   // one float4 per thread
    if (t < n_nodes * 4) {
        int i = t >> 2;          // node index
        int q = t & 3;           // which quarter of the 16-wide feature row
        float di = dinv[i];
        float s = di * di;
        float4 hv = h4[t];
        float4 bv = b4[q];
        float4 o;
        o.x = hv.x * s + bv.x;
        o.y = hv.y * s + bv.y;
        o.z = hv.z * s + bv.z;
        o.w = hv.w * s + bv.w;
        out4[t] = o;
    }
}

// ---------------------------------------------------------------------------
// 6) edge scatter: 16 consecutive lanes handle one edge's 16 features.
//    Gather h[row*16 + 0..15] is one coalesced 64B line; scatter-add to
//    out[col*16 + 0..15] is 16 f32 atomics into one 64B line.
// ---------------------------------------------------------------------------
__global__ void gcn_scatter(const long long* __restrict__ row,
                            const long long* __restrict__ col,
                            const float* __restrict__ h,
                            const float* __restrict__ dinv,
                            float* __restrict__ out, int n_edges) {
    long long t = (long long)blockIdx.x * blockDim.x + threadIdx.x;
    long long e = t >> 4;
    int k = (int)(t & 15);
    if (e >= n_edges) return;
    int r = (int)row[e];
    int c = (int)col[e];
    float norm = dinv[r] * dinv[c];
    atomicAdd(&out[(long long)c * 16 + k], h[(long long)r * 16 + k] * norm);
}

// ---------------------------------------------------------------------------
extern "C" void kernel_launch(void* const* d_in, const int* in_sizes, int n_in,
                              void* d_out, int out_size, void* d_ws, size_t ws_size,
                              hipStream_t stream) {
    const float*     x    = (const float*)d_in[0];      // [N,16]
    const long long* ei   = (const long long*)d_in[1];  // [2,E] int64
    const float*     W    = (const float*)d_in[2];      // [16,16]
    const float*     bias = (const float*)d_in[3];      // [16]
    float*           out  = (float*)d_out;

    const int n_nodes = in_sizes[0] / 16;
    const int n_edges = in_sizes[1] / 2;
    const long long* row = ei;            // edge_index[0]
    const long long* col = ei + n_edges;  // edge_index[1]

    // workspace layout: h [N*16 f32] | deg/dinv [N f32]
    float* h   = (float*)d_ws;
    float* deg = h + (size_t)n_nodes * 16;

    const int BS = 256;

    // 1) deg = 0
    gcn_zero_deg<<<(n_nodes + BS - 1) / BS, BS, 0, stream>>>(deg, n_nodes);

    // 2) deg[col[e]] += 1
    gcn_count_deg<<<(n_edges + BS - 1) / BS, BS, 0, stream>>>(col, deg, n_edges);

    // 3) h = x @ W  (WMMA; N/16 waves, 16 rows per wave)
    {
        long long threads = (long long)(((n_nodes + 15) / 16)) * 32;
        int blocks = (int)((threads + BS - 1) / BS);
        gcn_h_wmma<<<blocks, BS, 0, stream>>>(x, W, h, n_nodes);
    }

    // 4) dinv = rsqrt(deg + 1)
    gcn_dinv<<<(n_nodes + BS - 1) / BS, BS, 0, stream>>>(deg, n_nodes);

    // 5) out = h * dinv^2 + b  (float4 granularity)
    {
        int threads = n_nodes * 4;
        gcn_init_out<<<(threads + BS - 1) / BS, BS, 0, stream>>>(
            (const float4*)h, deg, (const float4*)bias, (float4*)out, n_nodes);
    }

    // 6) scatter messages over edges
    {
        long long threads = (long long)n_edges * 16;
        int blocks = (int)((threads + BS - 1) / BS);
        gcn_scatter<<<blocks, BS, 0, stream>>>(row, col, h, deg, out, n_edges);
    }
}